// GraphConvLayer_14972255993922
// MI455X (gfx1250) — compile-verified
//
#include <hip/hip_runtime.h>
#include <math.h>

typedef __attribute__((ext_vector_type(2))) float v2f;
typedef __attribute__((ext_vector_type(4))) float v4f;
typedef __attribute__((ext_vector_type(8))) float v8f;

#define NODES     50000
#define KNN       32
#define CDIM      256
#define M_TILE    16
#define THREADS   128
#define KCHUNK    16                 // K rows of W staged per chunk
#define PKCHUNK   (KCHUNK / 2)       // K-pairs per chunk
#define PSTRIDE   (2 * CDIM + 32)    // 544 words: pair rows shift banks by 32
#define MIXSTRIDE 520                // 512 + pad

__global__ __launch_bounds__(THREADS)
void graphconv_fused(const float* __restrict__ feat,
                     const float* __restrict__ coords,
                     const int*   __restrict__ knn,
                     const float* __restrict__ W,
                     const float* __restrict__ bias,
                     float* __restrict__ out)
{
    __shared__ float s_mix[M_TILE * MIXSTRIDE];  // [feat(256) | agg(256)] per node
    __shared__ float s_wp[PKCHUNK * PSTRIDE];    // W chunk, K-pair interleaved
    __shared__ float s_geo[M_TILE][8];           // rel_mean xyz, rel_std xyz
    __shared__ int   s_idx[M_TILE * KNN];

    const int tid   = threadIdx.x;
    const int node0 = blockIdx.x * M_TILE;

    // ---- Phase 1a: knn indices for this node tile (contiguous in memory) ----
    for (int i = tid; i < M_TILE * KNN; i += THREADS)
        s_idx[i] = knn[node0 * KNN + i];

    // ---- Phase 1b: copy feat tile -> mix[:, 0:256] (float4, coalesced) ----
    for (int i = tid; i < M_TILE * (CDIM / 4); i += THREADS) {
        const int r  = i / (CDIM / 4);
        const int c4 = (i % (CDIM / 4)) * 4;
        *(v4f*)&s_mix[r * MIXSTRIDE + c4] =
            *(const v4f*)&feat[(node0 + r) * CDIM + c4];
    }
    __syncthreads();

    // ---- Phase 1c: neighbor feature mean -> mix[:, 256:512] ----
    // Each thread owns 2 channels across all 16 nodes: lane-consecutive
    // channels => each neighbor row is one coalesced 1KB burst (L2-resident).
    {
        const int c0 = tid * 2;
        #pragma unroll 1
        for (int r = 0; r < M_TILE; ++r) {
            float a0 = 0.f, a1 = 0.f;
            #pragma unroll 4
            for (int j = 0; j < KNN; ++j) {
                const int nb = s_idx[r * KNN + j];
                const v2f v = *(const v2f*)&feat[nb * CDIM + c0];
                a0 += v.x; a1 += v.y;
            }
            s_mix[r * MIXSTRIDE + CDIM + c0]     = a0 * (1.0f / KNN);
            s_mix[r * MIXSTRIDE + CDIM + c0 + 1] = a1 * (1.0f / KNN);
        }
    }

    // ---- Phase 1d: relative-coordinate mean / population std ----
    if (tid < M_TILE) {
        const int r = tid;
        const float cx = coords[(node0 + r) * 3 + 0];
        const float cy = coords[(node0 + r) * 3 + 1];
        const float cz = coords[(node0 + r) * 3 + 2];
        float sx = 0.f, sy = 0.f, sz = 0.f, qx = 0.f, qy = 0.f, qz = 0.f;
        #pragma unroll 4
        for (int j = 0; j < KNN; ++j) {
            const int nb = s_idx[r * KNN + j];
            const float dx = coords[nb * 3 + 0] - cx;
            const float dy = coords[nb * 3 + 1] - cy;
            const float dz = coords[nb * 3 + 2] - cz;
            sx += dx; sy += dy; sz += dz;
            qx += dx * dx; qy += dy * dy; qz += dz * dz;
        }
        const float inv = 1.0f / KNN;
        const float mx = sx * inv, my = sy * inv, mz = sz * inv;
        s_geo[r][0] = mx;
        s_geo[r][1] = my;
        s_geo[r][2] = mz;
        s_geo[r][3] = sqrtf(fmaxf(qx * inv - mx * mx, 0.f));
        s_geo[r][4] = sqrtf(fmaxf(qy * inv - my * my, 0.f));
        s_geo[r][5] = sqrtf(fmaxf(qz * inv - mz * mz, 0.f));
        s_geo[r][6] = 0.f;
        s_geo[r][7] = 0.f;
    }

    // ---- Phase 2: GEMM  out_tile = mix(16x512) @ W[0:512,:] via f32 WMMA ----
    const int wv   = tid >> 5;        // wave id 0..3 -> columns wv*64..wv*64+63
    const int lane = tid & 31;
    const int lg   = lane >> 4;       // lane group (0: lanes 0-15, 1: 16-31)
    const int ln   = lane & 15;

    v8f acc[4] = {};                  // 4 N-tiles of 16x16 f32 accumulators

    for (int k0 = 0; k0 < 2 * CDIM; k0 += KCHUNK) {
        __syncthreads();              // previous s_wp chunk fully consumed
        // Stage W[k0 .. k0+15][:] as interleaved K-pairs:
        //   s_wp[p][2n+0] = W[k0+2p  ][n]
        //   s_wp[p][2n+1] = W[k0+2p+1][n]
        // => one aligned ds_load_b64 per B fragment, no VGPR repacking.
        for (int i = tid; i < PKCHUNK * (CDIM / 4); i += THREADS) {
            const int p  = i / (CDIM / 4);
            const int c4 = (i % (CDIM / 4)) * 4;
            const v4f w0 = *(const v4f*)&W[(k0 + 2 * p    ) * CDIM + c4];
            const v4f w1 = *(const v4f*)&W[(k0 + 2 * p + 1) * CDIM + c4];
            float* d = &s_wp[p * PSTRIDE + 2 * c4];
            v2f p0 = {w0.x, w1.x}; *(v2f*)&d[0] = p0;
            v2f p1 = {w0.y, w1.y}; *(v2f*)&d[2] = p1;
            v2f p2 = {w0.z, w1.z}; *(v2f*)&d[4] = p2;
            v2f p3 = {w0.w, w1.w}; *(v2f*)&d[6] = p3;
        }
        __syncthreads();

        #pragma unroll
        for (int kk = 0; kk < KCHUNK; kk += 4) {
            // A fragment (16x4): lane holds row M=ln, K = kk + 2*lg + {0,1}
            const v2f a = *(const v2f*)&s_mix[ln * MIXSTRIDE + k0 + kk + 2 * lg];
            // B fragment (4x16): needs K rows kk+2*lg, kk+2*lg+1  => pair
            const int p = (kk >> 1) + lg;
            #pragma unroll
            for (int t = 0; t < 4; ++t) {
                const int n = wv * 64 + t * 16 + ln;
                const v2f bf = *(const v2f*)&s_wp[p * PSTRIDE + 2 * n];
                acc[t] = __builtin_amdgcn_wmma_f32_16x16x4_f32(
                    false, a, false, bf, (short)0, acc[t], false, false);
            }
        }
    }

    // ---- Epilogue: + geo @ W[512:518,:] + bias, SiLU, store ----
    __syncthreads();
    #pragma unroll
    for (int t = 0; t < 4; ++t) {
        const int n = wv * 64 + t * 16 + ln;
        const float bn = bias[n];
        float wg[6];
        #pragma unroll
        for (int j = 0; j < 6; ++j)
            wg[j] = W[(2 * CDIM + j) * CDIM + n];
        #pragma unroll
        for (int r = 0; r < 8; ++r) {
            const int m = r + 8 * lg;   // C/D layout: VGPR r -> rows r, r+8
            float v = acc[t][r] + bn;
            #pragma unroll
            for (int j = 0; j < 6; ++j)
                v += s_geo[m][j] * wg[j];
            const float sig = 1.0f / (1.0f + __expf(-v));
            out[(node0 + m) * CDIM + n] = v * sig;
        }
    }
}

extern "C" void kernel_launch(void* const* d_in, const int* in_sizes, int n_in,
                              void* d_out, int out_size, void* d_ws, size_t ws_size,
                              hipStream_t stream) {
    const float* feat   = (const float*)d_in[0];
    const float* coords = (const float*)d_in[1];
    const int*   knn    = (const int*)d_in[2];
    const float* W      = (const float*)d_in[3];
    const float* bias   = (const float*)d_in[4];
    float* out = (float*)d_out;

    dim3 grid(NODES / M_TILE);   // 50000 / 16 = 3125 exactly
    graphconv_fused<<<grid, THREADS, 0, stream>>>(feat, coords, knn, W, bias, out);
}